// DegreeOnlyFiltration_23665269801452
// MI455X (gfx1250) — compile-verified
//
#include <hip/hip_runtime.h>
#include <hip/hip_bf16.h>
#include <stdint.h>

#define TPB        256      // 8 wave32 waves per workgroup
#define NWAVES     (TPB / 32)
#define LDS_ELEMS  16384    // 64 KB f32 staging tile -> 5 WGs per 320KB WGP

typedef unsigned int u32x4 __attribute__((ext_vector_type(4)));
typedef int          i32x4 __attribute__((ext_vector_type(4)));
typedef int          i32x8 __attribute__((ext_vector_type(8)));

// ---- CDNA5 async global->LDS copy path (ASYNCcnt tracked) ------------------
__device__ __forceinline__ void async_ld_b128(unsigned lds_off, const void* gaddr) {
    asm volatile("global_load_async_to_lds_b128 %0, %1, off"
                 :: "v"(lds_off), "v"(gaddr) : "memory");
}
__device__ __forceinline__ void async_ld_b32(unsigned lds_off, const void* gaddr) {
    asm volatile("global_load_async_to_lds_b32 %0, %1, off"
                 :: "v"(lds_off), "v"(gaddr) : "memory");
}
__device__ __forceinline__ void wait_async0() {
    asm volatile("s_wait_asynccnt 0" ::: "memory");
}

// ---- CDNA5 Tensor Data Mover: 1-row 2D tile DMA into LDS -------------------
#if __has_builtin(__builtin_amdgcn_tensor_load_to_lds)
#define HAVE_TDM 1
__device__ __forceinline__ void tdm_row_to_lds(unsigned lds_off,
                                               const float* gsrc, int elems) {
    const unsigned long long ga = (unsigned long long)(uintptr_t)gsrc;
    // D# group 0: count=1 (valid), lds_addr, 57-bit global_addr, type=2 ("image")
    u32x4 g0 = { 1u,
                 lds_off,
                 (unsigned)ga,
                 (unsigned)((ga >> 32) & 0x01FFFFFFu) | 0x80000000u };
    const unsigned n = (unsigned)elems;
    // D# group 1: mask=0, data_size=2 (4B), no barrier/iterate/pad,
    //   tensor_dim0 = n, tensor_dim1 = 1, tile_dim0 = n, tile_dim1 = 1,
    //   tensor_dim0_stride = n, tensor_dim1_stride = 0
    i32x8 g1 = { (int)0x00020000u,                       // data_size=4B
                 (int)((n & 0xFFFFu) << 16),             // tensor_dim0 lo16
                 (int)((n >> 16) | (1u << 16)),          // tensor_dim0 hi16 | tensor_dim1 lo16 (=1)
                 (int)((n & 0xFFFFu) << 16),             // tensor_dim1 hi16 (=0) | tile_dim0
                 1,                                      // tile_dim1 = 1, tile_dim2 = 0
                 (int)n,                                 // tensor_dim0_stride lo32
                 0, 0 };                                 // stride hi / dim1_stride
    i32x4 gz = { 0, 0, 0, 0 };                           // groups 2/3 unused (<=2D)
#if __clang_major__ >= 23
    i32x8 gz8 = { 0, 0, 0, 0, 0, 0, 0, 0 };
    __builtin_amdgcn_tensor_load_to_lds(g0, g1, gz, gz, gz8, 0);
#else
    __builtin_amdgcn_tensor_load_to_lds(g0, g1, gz, gz, 0);
#endif
}
#else
#define HAVE_TDM 0
#endif

// ---- wave32 max reduction ---------------------------------------------------
__device__ __forceinline__ float wave_max32(float v) {
    #pragma unroll
    for (int off = 16; off >= 1; off >>= 1)
        v = fmaxf(v, __shfl_xor(v, off, 32));
    return v;
}

__global__ __launch_bounds__(TPB)
void seg_max_norm_kernel(const float* __restrict__ deg,
                         const int*   __restrict__ pos,
                         float*       __restrict__ out) {
    __shared__ float s_tile[LDS_ELEMS];
    __shared__ float s_red[NWAVES];

    const int g   = blockIdx.x;
    const int beg = pos[g];
    const int end = pos[g + 1];
    const int len = end - beg;
    if (len <= 0) return;

    const int tid  = threadIdx.x;
    const int lane = tid & 31;
    const int wid  = tid >> 5;

    const float* src = deg + beg;
    float*       dst = out + beg;

    const float NEG_INF = -__builtin_inff();

    // Fast path: segment fits in the LDS tile and is 16B aligned -> single HBM read.
    const bool fast = (len <= LDS_ELEMS) && ((((uintptr_t)src) & 15u) == 0);

    float m = NEG_INF;
    const int n4 = len >> 2;   // float4 chunks

    if (fast) {
        const unsigned lds_base = (unsigned)(uintptr_t)(&s_tile[0]);
#if HAVE_TDM
        // Each wave DMAs its contiguous slice of the segment via the TDM.
        {
            const int chunk = (len + NWAVES - 1) / NWAVES;
            const int off   = wid * chunk;
            const int myLen = min(len - off, chunk);
            if (myLen > 0)
                tdm_row_to_lds(lds_base + 4u * (unsigned)off, src + off, myLen);
            __builtin_amdgcn_s_wait_tensorcnt(0);   // each wave drains its own DMA
        }
#else
        // Fallback: per-lane async b128 copies (ASYNCcnt tracked).
        for (int i = tid; i < n4; i += TPB)
            async_ld_b128(lds_base + 16u * (unsigned)i, (const void*)(src + 4 * i));
        for (int i = (n4 << 2) + tid; i < len; i += TPB)   // 0..3 tail elements
            async_ld_b32(lds_base + 4u * (unsigned)i, (const void*)(src + i));
        wait_async0();
#endif
        __syncthreads();      // tile visible to all 8 waves

        const float4* t4 = (const float4*)s_tile;
        for (int i = tid; i < n4; i += TPB) {
            float4 v = t4[i];
            m = fmaxf(m, fmaxf(fmaxf(v.x, v.y), fmaxf(v.z, v.w)));
        }
        for (int i = (n4 << 2) + tid; i < len; i += TPB)
            m = fmaxf(m, s_tile[i]);
    } else {
        // Generic fallback: two passes over global (2nd pass hits 192MB L2 anyway).
        for (int i = tid; i < len; i += TPB)
            m = fmaxf(m, src[i]);
    }

    // Block reduction: wave32 shuffle tree, then combine the 8 wave maxima.
    m = wave_max32(m);
    if (lane == 0) s_red[wid] = m;
    __syncthreads();
    if (wid == 0) {
        float r = (lane < NWAVES) ? s_red[lane] : NEG_INF;
        r = wave_max32(r);
        if (lane == 0) s_red[0] = r;
    }
    __syncthreads();

    const float inv = 1.0f / s_red[0];   // one precise divide per block (uniform)

    if (fast) {
        const float4* t4 = (const float4*)s_tile;
        float4*       o4 = (float4*)dst;
        for (int i = tid; i < n4; i += TPB) {
            float4 v = t4[i];
            v.x *= inv; v.y *= inv; v.z *= inv; v.w *= inv;
            o4[i] = v;                   // b128 store straight to HBM
        }
        for (int i = (n4 << 2) + tid; i < len; i += TPB)
            dst[i] = s_tile[i] * inv;
    } else {
        for (int i = tid; i < len; i += TPB)
            dst[i] = src[i] * inv;
    }
}

extern "C" void kernel_launch(void* const* d_in, const int* in_sizes, int n_in,
                              void* d_out, int out_size, void* d_ws, size_t ws_size,
                              hipStream_t stream) {
    const float* deg = (const float*)d_in[0];   // node_deg, float32 [N]
    const int*   pos = (const int*)d_in[1];     // sample_pos, int32 [B+1]
    float*       out = (float*)d_out;           // float32 [N]

    const int n_graphs = in_sizes[1] - 1;       // 4096 segments
    seg_max_norm_kernel<<<n_graphs, TPB, 0, stream>>>(deg, pos, out);
}